// AttnPooling_12695923327590
// MI455X (gfx1250) — compile-verified
//
#include <hip/hip_runtime.h>
#include <hip/hip_bf16.h>

typedef __attribute__((ext_vector_type(2))) float v2f;
typedef __attribute__((ext_vector_type(8))) float v8f;

#define T_DIM 4096
#define DIN_DIM 512
#define DH_DIM 500
#define B_DIM 32
#define TT 64      // t-values per workgroup
#define KP 32      // k-panel depth (16 k-pairs)
#define XSP 160    // x tile pair-row stride (floats): 160%64=32 -> lane halves on disjoint banks
#define WSP 1056   // W1 panel pair-row stride (floats): 1056%64=32 -> disjoint banks

// ---------------------------------------------------------------------------
// Kernel 1: scores[b,t] = w2 . relu( W1 @ x[b,:,t] ), via V_WMMA_F32_16X16X4_F32
// LDS panels stored k-pair-interleaved: fragment = one contiguous ds_load_b64
// ---------------------------------------------------------------------------
__global__ __launch_bounds__(512)
void attn_scores_kernel(const float* __restrict__ x,
                        const float* __restrict__ w1,
                        const float* __restrict__ w2,
                        float* __restrict__ scores) {
    __shared__ float xs2[(KP / 2) * XSP];   // [kpair][tt*2 + k&1]   10.0 KB
    __shared__ float ws2[(KP / 2) * WSP];   // [kpair][h*2  + k&1]   67.6 KB (h zero-padded to 512)
    __shared__ float sPart[8][64];          // per-h-block partial scores

    const int tid = threadIdx.x;
    const int b  = blockIdx.y;
    const int t0 = blockIdx.x * TT;

    const int wave = tid >> 5;
    const int lane = tid & 31;
    const int s2 = wave & 1;             // t-half (32 t)
    const int q8 = wave >> 1;            // h-block (64 h), 8 blocks cover padded 512
    const int n  = lane & 15;            // N (B/C/D) or M (A) index
    const int hh = lane >> 4;            // lane half -> k-pair row offset

    const float* xb = x + (size_t)b * DIN_DIM * T_DIM;

    v8f acc[2][4];
    const v8f zero8 = {0.f,0.f,0.f,0.f,0.f,0.f,0.f,0.f};
#pragma unroll
    for (int st = 0; st < 2; ++st)
#pragma unroll
        for (int ht = 0; ht < 4; ++ht) acc[st][ht] = zero8;

    // cooperative-load index helpers
    const int xkk = tid >> 4;            // 0..31 (k row of x tile)
    const int xtt = (tid & 15) * 4;      // 0..60 (t col, float4)
    const int wh  = tid & 63;            // h within 64-group
    const int wkg = tid >> 6;            // 0..7, each covers 4 kk (2 k-pairs)

    for (int kp = 0; kp < DIN_DIM / KP; ++kp) {
        const int k0 = kp * KP;
        __syncthreads();
        // x tile: xs2[(kk>>1)*XSP + tt*2 + (kk&1)] = x[b][k0+kk][t0+tt]
        {
            const float4 v = *(const float4*)(xb + (size_t)(k0 + xkk) * T_DIM + t0 + xtt);
            float* dst = xs2 + (xkk >> 1) * XSP + xtt * 2 + (xkk & 1);
            dst[0] = v.x; dst[2] = v.y; dst[4] = v.z; dst[6] = v.w;
        }
        // W1 panel (transpose + zero-pad + pair-interleave):
        // ws2[(kk>>1)*WSP + h*2 + (kk&1)] = (h<500) ? w1[h][k0+kk] : 0
#pragma unroll
        for (int r = 0; r < 8; ++r) {
            const int h = r * 64 + wh;
            float4 v = {0.f, 0.f, 0.f, 0.f};
            if (h < DH_DIM) v = *(const float4*)(w1 + (size_t)h * DIN_DIM + k0 + wkg * 4);
            v2f p0 = {v.x, v.y};         // k-pair (4g, 4g+1)
            v2f p1 = {v.z, v.w};         // k-pair (4g+2, 4g+3)
            *(v2f*)(ws2 + (wkg * 2 + 0) * WSP + h * 2) = p0;
            *(v2f*)(ws2 + (wkg * 2 + 1) * WSP + h * 2) = p1;
        }
        // prefetch next panel into L2/L0 while we compute this one
        if (kp + 1 < DIN_DIM / KP) {
            __builtin_prefetch(xb + (size_t)(k0 + KP + xkk) * T_DIM + t0 + xtt, 0, 3);
            __builtin_prefetch(w1 + (size_t)wh * DIN_DIM + k0 + KP + wkg * 4, 0, 3);
        }
        __syncthreads();

#pragma unroll
        for (int ks = 0; ks < 8; ++ks) {
            const int pk = ks * 2 + hh;              // k-pair row; lane half picks k = ks*4+2*hh(+j)
            v2f a[2];                                 // A[m=t][k], m = n
#pragma unroll
            for (int st = 0; st < 2; ++st)
                a[st] = *(const v2f*)(xs2 + pk * XSP + (s2 * 32 + st * 16 + n) * 2);
#pragma unroll
            for (int ht = 0; ht < 4; ++ht) {
                const v2f bf = *(const v2f*)(ws2 + pk * WSP + (q8 * 64 + ht * 16 + n) * 2);
#pragma unroll
                for (int st = 0; st < 2; ++st)
                    acc[st][ht] = __builtin_amdgcn_wmma_f32_16x16x4_f32(
                        false, a[st], false, bf, (short)0, acc[st][ht], false, false);
            }
        }
    }

    // relu + dot with w2 along this wave's 64 h-columns
    float w2v[4];
#pragma unroll
    for (int ht = 0; ht < 4; ++ht) {
        const int h = q8 * 64 + ht * 16 + n;
        w2v[ht] = (h < DH_DIM) ? w2[h] : 0.f;
    }
#pragma unroll
    for (int st = 0; st < 2; ++st) {
        float part[8];
#pragma unroll
        for (int j = 0; j < 8; ++j) part[j] = 0.f;
#pragma unroll
        for (int ht = 0; ht < 4; ++ht) {
#pragma unroll
            for (int j = 0; j < 8; ++j) {
                float v = acc[st][ht][j];             // (M = j + 8*hh, N = n)
                v = v > 0.f ? v : 0.f;                // relu
                part[j] += v * w2v[ht];
            }
        }
        // reduce over the 16 lanes of each half (xor masks < 16 stay within half)
#pragma unroll
        for (int j = 0; j < 8; ++j) {
#pragma unroll
            for (int off = 8; off >= 1; off >>= 1)
                part[j] += __shfl_xor(part[j], off, 32);
        }
        if (n == 0) {
#pragma unroll
            for (int j = 0; j < 8; ++j)
                sPart[q8][s2 * 32 + st * 16 + j + 8 * hh] = part[j];
        }
    }
    __syncthreads();
    if (tid < 64) {                                   // deterministic fixed-order combine
        float sc = 0.f;
#pragma unroll
        for (int qq = 0; qq < 8; ++qq) sc += sPart[qq][tid];
        scores[(size_t)b * T_DIM + t0 + tid] = sc;
    }
}

// ---------------------------------------------------------------------------
// Kernel 2: per-batch softmax stats (max, 1/sum-exp)
// ---------------------------------------------------------------------------
__global__ __launch_bounds__(256)
void softmax_stats_kernel(const float* __restrict__ scores, float* __restrict__ stats) {
    __shared__ float red[256];
    const int b = blockIdx.x;
    const int tid = threadIdx.x;
    const float* sc = scores + (size_t)b * T_DIM;

    float m = -3.402823466e38f;
    for (int t = tid; t < T_DIM; t += 256) m = fmaxf(m, sc[t]);
    red[tid] = m; __syncthreads();
    for (int o = 128; o > 0; o >>= 1) {
        if (tid < o) red[tid] = fmaxf(red[tid], red[tid + o]);
        __syncthreads();
    }
    const float smax = red[0];
    __syncthreads();

    float ssum = 0.f;
    for (int t = tid; t < T_DIM; t += 256) ssum += __expf(sc[t] - smax);
    red[tid] = ssum; __syncthreads();
    for (int o = 128; o > 0; o >>= 1) {
        if (tid < o) red[tid] += red[tid + o];
        __syncthreads();
    }
    if (tid == 0) { stats[b * 2] = smax; stats[b * 2 + 1] = 1.f / red[0]; }
}

// ---------------------------------------------------------------------------
// Kernel 3: mean = x @ attn, var = E[x^2] - 2 m E[x] + m^2, out = [mean | std]
// ---------------------------------------------------------------------------
__global__ __launch_bounds__(256)
void pool_stats_kernel(const float* __restrict__ x, const float* __restrict__ scores,
                       const float* __restrict__ stats, float* __restrict__ out) {
    __shared__ float attn[T_DIM];                     // 16 KB normalized weights
    const int b  = blockIdx.y;
    const int d0 = blockIdx.x * 16;
    const int tid = threadIdx.x;

    const float smax = stats[b * 2];
    const float inv  = stats[b * 2 + 1];
    const float* sc  = scores + (size_t)b * T_DIM;
    for (int t = tid; t < T_DIM; t += 256) attn[t] = __expf(sc[t] - smax) * inv;
    __syncthreads();

    const int d = d0 + (tid >> 4);
    const int l = tid & 15;                           // 16 lanes cooperate per d-row
    const float4* xr = (const float4*)(x + ((size_t)b * DIN_DIM + d) * T_DIM);
    const float4* aw = (const float4*)attn;

    float wsum = 0.f, s1 = 0.f, s2 = 0.f;
#pragma unroll 4
    for (int i2 = 0; i2 < T_DIM / 4 / 16; ++i2) {
        const int i = i2 * 16 + l;                    // consecutive lanes -> consecutive float4
        const float4 v = xr[i];
        const float4 w = aw[i];
        wsum += v.x * w.x + v.y * w.y + v.z * w.z + v.w * w.w;
        s1   += v.x + v.y + v.z + v.w;
        s2   += v.x * v.x + v.y * v.y + v.z * v.z + v.w * v.w;
    }
#pragma unroll
    for (int off = 8; off >= 1; off >>= 1) {
        wsum += __shfl_xor(wsum, off, 32);
        s1   += __shfl_xor(s1, off, 32);
        s2   += __shfl_xor(s2, off, 32);
    }
    if (l == 0) {
        const float invT = 1.f / (float)T_DIM;
        const float mean = wsum;
        const float e1 = s1 * invT, e2 = s2 * invT;
        float var = e2 - 2.f * mean * e1 + mean * mean;
        if (var <= 1e-12f) var = 1e-12f;
        out[(size_t)b * (2 * DIN_DIM) + d]            = mean;
        out[(size_t)b * (2 * DIN_DIM) + DIN_DIM + d]  = sqrtf(var);
    }
}

// ---------------------------------------------------------------------------
extern "C" void kernel_launch(void* const* d_in, const int* in_sizes, int n_in,
                              void* d_out, int out_size, void* d_ws, size_t ws_size,
                              hipStream_t stream) {
    const float* x  = (const float*)d_in[0];   // (32, 512, 4096)
    const float* w1 = (const float*)d_in[1];   // (500, 512)
    const float* w2 = (const float*)d_in[2];   // (1, 500)
    // d_in[3] = dim (==2), compile-time constant here
    float* out    = (float*)d_out;             // (32, 1024)
    float* scores = (float*)d_ws;              // B*T floats = 512 KB
    float* stats  = scores + (size_t)B_DIM * T_DIM;  // 2 floats per batch

    attn_scores_kernel<<<dim3(T_DIM / TT, B_DIM), 512, 0, stream>>>(x, w1, w2, scores);
    softmax_stats_kernel<<<B_DIM, 256, 0, stream>>>(scores, stats);
    pool_stats_kernel<<<dim3(DIN_DIM / 16, B_DIM), 256, 0, stream>>>(x, scores, stats, out);
}